// Aggregator_55018531062593
// MI455X (gfx1250) — compile-verified
//
#include <hip/hip_runtime.h>
#include <hip/hip_bf16.h>
#include <cmath>

typedef __attribute__((ext_vector_type(2))) float v2f;
typedef __attribute__((ext_vector_type(8))) float v8f;

#define CHANNEL 64
#define NFACT   4
#define NREL    16

// ---------------------------------------------------------------- zero fill
__global__ void zero_kernel(float* __restrict__ p, long long n) {
  long long i = (long long)blockIdx.x * blockDim.x + threadIdx.x;
  long long stride = (long long)gridDim.x * blockDim.x;
  for (; i < n; i += stride) p[i] = 0.0f;
}

// ------------------------------------------- disen_weight = softmax(att)@W
// 1 block, 64 threads: thread c computes dw[f][c] for all 4 factors.
__global__ void dw_kernel(const float* __restrict__ att,     // [4,16]
                          const float* __restrict__ weight,  // [16,64]
                          float* __restrict__ dw) {          // [4,64]
  int c = threadIdx.x;  // 0..63
  for (int f = 0; f < NFACT; ++f) {
    float mx = -1e30f;
    for (int r = 0; r < NREL; ++r) mx = fmaxf(mx, att[f * NREL + r]);
    float e[NREL], s = 0.0f;
    for (int r = 0; r < NREL; ++r) { e[r] = __expf(att[f * NREL + r] - mx); s += e[r]; }
    float inv = 1.0f / s;
    float acc = 0.0f;
    for (int r = 0; r < NREL; ++r) acc += (e[r] * inv) * weight[r * CHANNEL + c];
    dw[f * CHANNEL + c] = acc;
  }
}

// ------------------------------------------------------ KG edge scatter-add
// One wave (32 lanes) per edge; lane l handles channels 2l, 2l+1.
__global__ void kg_scatter_kernel(const float* __restrict__ entity_emb,
                                  const int* __restrict__ head,
                                  const int* __restrict__ tail,
                                  const int* __restrict__ etype,
                                  const float* __restrict__ kg_mask,
                                  const float* __restrict__ weight,
                                  float* __restrict__ entity_out,   // [N_E,64]
                                  float* __restrict__ counts,       // [N_E]
                                  int E) {
  int tid = blockIdx.x * blockDim.x + threadIdx.x;
  int e = tid >> 5;
  if (e >= E) return;
  int lane = tid & 31;
  int h = head[e];
  int t = tail[e];
  int r = etype[e] - 1;
  float km = kg_mask[e];
  const float2* emb = (const float2*)(entity_emb + (size_t)t * CHANNEL);
  const float2* wr  = (const float2*)(weight + (size_t)r * CHANNEL);
  float2 v  = emb[lane];
  float2 ww = wr[lane];
  float* dst = entity_out + (size_t)h * CHANNEL + lane * 2;
  unsafeAtomicAdd(dst + 0, v.x * ww.x * km);
  unsafeAtomicAdd(dst + 1, v.y * ww.y * km);
  if (lane == 0) unsafeAtomicAdd(&counts[h], 1.0f);
}

// --------------------------------------------------- scatter_mean divide
__global__ void entity_div_kernel(float* __restrict__ entity_out,
                                  const float* __restrict__ counts, int n) {
  int tid = blockIdx.x * blockDim.x + threadIdx.x;
  int i = tid >> 5;
  if (i >= n) return;
  int lane = tid & 31;
  float inv = 1.0f / fmaxf(counts[i], 1.0f);
  float2* p = (float2*)(entity_out + (size_t)i * CHANNEL) + lane;
  float2 v = *p;
  v.x *= inv; v.y *= inv;
  *p = v;
}

// ------------------------------------------------- user sparse scatter-add
__global__ void user_scatter_kernel(const float* __restrict__ entity_emb,
                                    const int* __restrict__ rows,
                                    const int* __restrict__ cols,
                                    const float* __restrict__ vals,
                                    float* __restrict__ user_out,  // [N_U,64]
                                    int NNZ) {
  int tid = blockIdx.x * blockDim.x + threadIdx.x;
  int e = tid >> 5;
  if (e >= NNZ) return;
  int lane = tid & 31;
  int rr = rows[e];
  int cc = cols[e];
  float vv = vals[e];
  const float2* emb = (const float2*)(entity_emb + (size_t)cc * CHANNEL);
  float2 v = emb[lane];
  float* dst = user_out + (size_t)rr * CHANNEL + lane * 2;
  unsafeAtomicAdd(dst + 0, v.x * vv);
  unsafeAtomicAdd(dst + 1, v.y * vv);
}

// ---------------------------------------------------------- WMMA finalize
// One wave handles 16 users. Lane L: m = L%16 (user row), hi = L/16.
// Lane computes logits for factors (2*hi, 2*hi+1) -> 4-way softmax via
// shfl_xor(16) -> scores land exactly in the V_WMMA_F32_16X16X4_F32
// A-operand layout (16x4 f32 A: lane L holds M=L%16, K=2*hi+j).
// Then 4 WMMAs (one per 16-channel tile) compute mix = score @ dw, and
// user_out *= (1 + mix).  Launched with blockDim=32, grid = U/16 exactly:
// EXEC is all-1s everywhere (WMMA requirement), no divergent exits.
__global__ __launch_bounds__(32) void user_finalize_wmma_kernel(
    const float* __restrict__ user_emb,    // [U,64]
    const float* __restrict__ latent_emb,  // [4,64]
    const float* __restrict__ dw,          // [4,64]
    float* __restrict__ user_out) {        // [U,64]
  int wave = blockIdx.x;
  int lane = threadIdx.x;      // 0..31
  int m  = lane & 15;
  int hi = lane >> 4;          // 0 or 1
  int f0 = hi * 2;             // this lane's factor pair

  int u0 = wave * 16;
  const float* urow = user_emb + (size_t)(u0 + m) * CHANNEL;
  const float* lr0  = latent_emb + (size_t)f0 * CHANNEL;
  const float* lr1  = latent_emb + (size_t)(f0 + 1) * CHANNEL;

  float l0 = 0.0f, l1 = 0.0f;
#pragma unroll
  for (int k = 0; k < CHANNEL; k += 4) {
    float4 uu = *(const float4*)(urow + k);
    float4 a0 = *(const float4*)(lr0 + k);
    float4 a1 = *(const float4*)(lr1 + k);
    l0 = fmaf(uu.x, a0.x, fmaf(uu.y, a0.y, fmaf(uu.z, a0.z, fmaf(uu.w, a0.w, l0))));
    l1 = fmaf(uu.x, a1.x, fmaf(uu.y, a1.y, fmaf(uu.z, a1.z, fmaf(uu.w, a1.w, l1))));
  }
  // Exchange the other factor pair for the same user (lane ^ 16).
  float o0 = __shfl_xor(l0, 16, 32);
  float o1 = __shfl_xor(l1, 16, 32);
  float mx = fmaxf(fmaxf(l0, l1), fmaxf(o0, o1));
  float e0 = __expf(l0 - mx), e1 = __expf(l1 - mx);
  float denom = e0 + e1 + __expf(o0 - mx) + __expf(o1 - mx);
  float inv = 1.0f / denom;

  // A operand: A[m][2*hi + j] = score(user m, factor 2*hi + j)
  v2f A; A.x = e0 * inv; A.y = e1 * inv;

#pragma unroll
  for (int nc = 0; nc < 4; ++nc) {
    // B operand (4x16, K=4): VGPR j, lane L -> B[k = j + 2*hi][n = m]
    v2f B;
    B.x = dw[(0 + 2 * hi) * CHANNEL + nc * 16 + m];
    B.y = dw[(1 + 2 * hi) * CHANNEL + nc * 16 + m];
    v8f C = {};
    C = __builtin_amdgcn_wmma_f32_16x16x4_f32(
        /*neg_a=*/false, A, /*neg_b=*/false, B,
        /*c_mod=*/(short)0, C, /*reuse_a=*/false, /*reuse_b=*/false);
    // C/D layout: VGPR i, lane L -> mix[M = i + 8*hi][N = m]
#pragma unroll
    for (int i = 0; i < 8; ++i) {
      int row = u0 + i + 8 * hi;
      float* p = user_out + (size_t)row * CHANNEL + nc * 16 + m;
      *p = *p * (1.0f + C[i]);
    }
  }
}

// Scalar tail for user counts not divisible by 16 (unused for U=50000).
__global__ void user_finalize_tail_kernel(const float* __restrict__ user_emb,
                                          const float* __restrict__ latent_emb,
                                          const float* __restrict__ dw,
                                          float* __restrict__ user_out,
                                          int ustart, int U) {
  int u = ustart + blockIdx.x;
  if (u >= U) return;
  int c = threadIdx.x;  // 64 threads
  float logits[NFACT];
  for (int f = 0; f < NFACT; ++f) {
    float acc = 0.0f;
    for (int k = 0; k < CHANNEL; ++k)
      acc += user_emb[(size_t)u * CHANNEL + k] * latent_emb[f * CHANNEL + k];
    logits[f] = acc;
  }
  float mx = fmaxf(fmaxf(logits[0], logits[1]), fmaxf(logits[2], logits[3]));
  float s = 0.0f;
  for (int f = 0; f < NFACT; ++f) { logits[f] = __expf(logits[f] - mx); s += logits[f]; }
  float mixv = 0.0f;
  for (int f = 0; f < NFACT; ++f) mixv += (logits[f] / s) * dw[f * CHANNEL + c];
  float* p = user_out + (size_t)u * CHANNEL + c;
  *p = *p * (1.0f + mixv);
}

extern "C" void kernel_launch(void* const* d_in, const int* in_sizes, int n_in,
                              void* d_out, int out_size, void* d_ws, size_t ws_size,
                              hipStream_t stream) {
  const float* entity_emb  = (const float*)d_in[0];
  const float* user_emb    = (const float*)d_in[1];
  const float* latent_emb  = (const float*)d_in[2];
  const int*   edge_index  = (const int*)d_in[3];   // [2,E]
  const int*   edge_type   = (const int*)d_in[4];   // [E]
  const int*   inter_idx   = (const int*)d_in[5];   // [2,NNZ]
  const float* inter_vals  = (const float*)d_in[6]; // [NNZ]
  const float* weight      = (const float*)d_in[7]; // [16,64]
  const float* disen_att   = (const float*)d_in[8]; // [4,16]
  const float* kg_mask     = (const float*)d_in[9]; // [E]

  const int E          = in_sizes[4];
  const int NNZ        = in_sizes[6];
  const int n_entities = in_sizes[0] / CHANNEL;
  const int n_users    = in_sizes[1] / CHANNEL;

  float* entity_out = (float*)d_out;                       // [n_entities,64]
  float* user_out   = entity_out + (size_t)n_entities * CHANNEL;  // [n_users,64]

  float* counts = (float*)d_ws;                            // [n_entities]
  float* dw     = counts + n_entities;                     // [4,64]

  const int* head = edge_index;
  const int* tail = edge_index + E;
  const int* rows = inter_idx;
  const int* cols = inter_idx + NNZ;

  // 1. zero output + counts
  zero_kernel<<<2048, 256, 0, stream>>>(entity_out, (long long)out_size);
  zero_kernel<<<512, 256, 0, stream>>>(counts, (long long)n_entities);

  // 2. disen_weight = softmax(att) @ weight
  dw_kernel<<<1, CHANNEL, 0, stream>>>(disen_att, weight, dw);

  // 3. KG scatter-add (1 wave per edge)
  {
    long long threads = (long long)E * 32;
    int blocks = (int)((threads + 255) / 256);
    kg_scatter_kernel<<<blocks, 256, 0, stream>>>(entity_emb, head, tail, edge_type,
                                                  kg_mask, weight, entity_out, counts, E);
  }

  // 4. scatter_mean divide
  {
    long long threads = (long long)n_entities * 32;
    int blocks = (int)((threads + 255) / 256);
    entity_div_kernel<<<blocks, 256, 0, stream>>>(entity_out, counts, n_entities);
  }

  // 5. user sparse scatter-add
  {
    long long threads = (long long)NNZ * 32;
    int blocks = (int)((threads + 255) / 256);
    user_scatter_kernel<<<blocks, 256, 0, stream>>>(entity_emb, rows, cols, inter_vals,
                                                    user_out, NNZ);
  }

  // 6. attention + mix via WMMA, apply user_agg *= (1 + mix)
  {
    int nwaves = n_users / 16;  // 3125 for U=50000 (exact)
    if (nwaves > 0)
      user_finalize_wmma_kernel<<<nwaves, 32, 0, stream>>>(user_emb, latent_emb, dw, user_out);
    int rem = n_users - nwaves * 16;
    if (rem > 0)
      user_finalize_tail_kernel<<<rem, CHANNEL, 0, stream>>>(user_emb, latent_emb, dw,
                                                             user_out, nwaves * 16, n_users);
  }
}